// GCN_58128087384142
// MI455X (gfx1250) — compile-verified
//
#include <hip/hip_runtime.h>
#include <hip/hip_bf16.h>

typedef float v2f __attribute__((ext_vector_type(2)));
typedef float v8f __attribute__((ext_vector_type(8)));

#define DIM 128

// ---------------------------------------------------------------- zero
__global__ void gcn_zero_kernel(float* __restrict__ p, long long n) {
    long long i = (long long)blockIdx.x * blockDim.x + threadIdx.x;
    if (i < n) p[i] = 0.0f;
}

// ---------------------------------------------------------------- degree count
__global__ void gcn_deg_kernel(const long long* __restrict__ dst,
                               float* __restrict__ deg, long long E) {
    long long t = (long long)blockIdx.x * blockDim.x + threadIdx.x;
    if (t < E) atomicAdd(&deg[dst[t]], 1.0f);
}

// deg -> rsqrt(deg + 1) in place
__global__ void gcn_dinv_kernel(float* __restrict__ deg, long long N) {
    long long i = (long long)blockIdx.x * blockDim.x + threadIdx.x;
    if (i < N) deg[i] = rsqrtf(deg[i] + 1.0f);
}

// ---------------------------------------------------------------- WMMA GEMM
// H[M,128] = X[M,128] @ W[128,128].  One block = 16 rows, 8 waves; each wave
// owns one 16x16 N-tile and walks K in steps of 4 with V_WMMA_F32_16X16X4_F32.
// M must be a multiple of 16 (100000 = 6250*16).
__global__ __launch_bounds__(256) void gcn_gemm_wmma_kernel(
        const float* __restrict__ X, const float* __restrict__ W,
        float* __restrict__ H, int M) {
    __shared__ float Wl[DIM * DIM];   // 64 KB of the 320 KB/WGP LDS

    // Cooperative 128x128 W load (float4 wide, fully coalesced)
    {
        const float4* Wv  = (const float4*)W;
        float4*       Wlv = (float4*)Wl;
        for (int i = threadIdx.x; i < DIM * DIM / 4; i += 256) Wlv[i] = Wv[i];
    }
    __syncthreads();

    const int wave = threadIdx.x >> 5;
    const int lane = threadIdx.x & 31;
    const int half = lane >> 4;        // K sub-phase: lanes 16-31 hold K+2,K+3
    const int l16  = lane & 15;

    const int row  = (blockIdx.x << 4) + l16;   // A-matrix M index for this lane
    const int col  = (wave << 4) + l16;         // B/D N index for this lane

    v8f c = {};
    const float* xrow = X + (size_t)row * DIM + half * 2;

#pragma unroll
    for (int k = 0; k < DIM; k += 4) {
        // A 16x4 fp32 fragment: VGPR0/1 = {K=k+2*half, K=k+1+2*half} per lane
        v2f a;
        {
            float2 t = *(const float2*)(xrow + k);
            a.x = t.x; a.y = t.y;
        }
        // B 4x16 fp32 fragment from LDS: rows k+2*half and k+1+2*half, col=lane&15
        v2f b;
        b.x = Wl[(k + half * 2) * DIM + col];
        b.y = Wl[(k + 1 + half * 2) * DIM + col];
        c = __builtin_amdgcn_wmma_f32_16x16x4_f32(
                /*neg_a=*/false, a, /*neg_b=*/false, b,
                /*c_mod=*/(short)0, c, /*reuse_a=*/false, /*reuse_b=*/false);
    }

    // D 16x16 fp32: VGPR r -> M = r + 8*half, N = col
    const int rbase = (blockIdx.x << 4) + half * 8;
    float* out = H + (size_t)rbase * DIM + col;
#pragma unroll
    for (int r = 0; r < 8; ++r) out[(size_t)r * DIM] = c[r];
}

// ---------------------------------------------------------------- edge scatter
// One wave per edge; each lane moves a float4 (32 lanes * 4 = 128 dims).
// Gather h[src] (L2-resident, 51.2MB < 192MB L2) and atomic-add into agg[dst].
__global__ __launch_bounds__(256) void gcn_scatter_kernel(
        const long long* __restrict__ src, const long long* __restrict__ dst,
        const float* __restrict__ dinv, const float* __restrict__ H,
        float* __restrict__ AGG, long long E) {
    long long t = (long long)blockIdx.x * blockDim.x + threadIdx.x;
    long long e = t >> 5;
    if (e >= E) return;
    const int d4 = (int)(t & 31) * 4;

    const long long s = src[e];
    const long long d = dst[e];
    const float norm = dinv[s] * dinv[d];

    const float4 hv = *(const float4*)&H[(size_t)s * DIM + d4];
    float* p = &AGG[(size_t)d * DIM + d4];
    atomicAdd(p + 0, hv.x * norm);
    atomicAdd(p + 1, hv.y * norm);
    atomicAdd(p + 2, hv.z * norm);
    atomicAdd(p + 3, hv.w * norm);
}

// ---------------------------------------------------------------- combine
// out = agg + h * dinv^2 + bias, optional ReLU.  One float4 per thread.
__global__ __launch_bounds__(256) void gcn_combine_kernel(
        const float* __restrict__ agg, const float* __restrict__ H,
        const float* __restrict__ dinv, const float* __restrict__ bias,
        float* __restrict__ out, long long N, int do_relu) {
    long long t = (long long)blockIdx.x * blockDim.x + threadIdx.x;
    if (t >= N * 32) return;
    const long long node = t >> 5;
    const int d4 = (int)(t & 31) * 4;

    const float di = dinv[node];
    const float sl = di * di;

    const float4 a  = *(const float4*)&agg[(size_t)node * DIM + d4];
    const float4 hv = *(const float4*)&H[(size_t)node * DIM + d4];
    const float4 bv = *(const float4*)&bias[d4];

    float4 r;
    r.x = fmaf(hv.x, sl, a.x) + bv.x;
    r.y = fmaf(hv.y, sl, a.y) + bv.y;
    r.z = fmaf(hv.z, sl, a.z) + bv.z;
    r.w = fmaf(hv.w, sl, a.w) + bv.w;
    if (do_relu) {
        r.x = fmaxf(r.x, 0.0f); r.y = fmaxf(r.y, 0.0f);
        r.z = fmaxf(r.z, 0.0f); r.w = fmaxf(r.w, 0.0f);
    }
    *(float4*)&out[(size_t)node * DIM + d4] = r;
}

// ---------------------------------------------------------------- launcher
extern "C" void kernel_launch(void* const* d_in, const int* in_sizes, int n_in,
                              void* d_out, int out_size, void* d_ws, size_t ws_size,
                              hipStream_t stream) {
    const float*     x  = (const float*)d_in[0];
    const long long* ei = (const long long*)d_in[1];   // int64 [2, E]
    const float*     W1 = (const float*)d_in[2];
    const float*     b1 = (const float*)d_in[3];
    const float*     W2 = (const float*)d_in[4];
    const float*     b2 = (const float*)d_in[5];

    const long long N = (long long)in_sizes[0] / DIM;   // 100000
    const long long E = (long long)in_sizes[1] / 2;     // 1600000
    const long long* src = ei;
    const long long* dst = ei + E;

    // Workspace carving: dinv (N f32) | bufH (N*128 f32) | bufB (N*128 f32)
    char* ws = (char*)d_ws;
    float* dinv = (float*)ws;
    size_t off1 = ((size_t)N * 4 + 511) & ~(size_t)511;
    float* bufH = (float*)(ws + off1);
    size_t off2 = off1 + (((size_t)N * DIM * 4 + 511) & ~(size_t)511);
    float* bufB = (float*)(ws + off2);
    float* agg  = (float*)d_out;   // d_out doubles as the aggregation buffer

    const int T = 256;
    const int bN    = (int)((N + T - 1) / T);
    const int bE    = (int)((E + T - 1) / T);
    const int bNF   = (int)((N * DIM + T - 1) / T);       // elementwise zero
    const int bE32  = (int)((E * 32 + T - 1) / T);        // scatter
    const int bN32  = (int)((N * 32 + T - 1) / T);        // combine
    const int bGemm = (int)(N / 16);                      // 6250 row-tiles

    // degrees -> deg_inv_sqrt
    gcn_zero_kernel<<<bN, T, 0, stream>>>(dinv, N);
    gcn_deg_kernel<<<bE, T, 0, stream>>>(dst, dinv, E);
    gcn_dinv_kernel<<<bN, T, 0, stream>>>(dinv, N);

    // ---- layer 1 ----
    gcn_gemm_wmma_kernel<<<bGemm, T, 0, stream>>>(x, W1, bufH, (int)N);
    gcn_zero_kernel<<<bNF, T, 0, stream>>>(agg, N * DIM);
    gcn_scatter_kernel<<<bE32, T, 0, stream>>>(src, dst, dinv, bufH, agg, E);
    gcn_combine_kernel<<<bN32, T, 0, stream>>>(agg, bufH, dinv, b1, bufB, N, /*relu=*/1);

    // ---- layer 2 ----
    gcn_gemm_wmma_kernel<<<bGemm, T, 0, stream>>>(bufB, W2, bufH, (int)N);
    gcn_zero_kernel<<<bNF, T, 0, stream>>>(agg, N * DIM);
    gcn_scatter_kernel<<<bE32, T, 0, stream>>>(src, dst, dinv, bufH, agg, E);
    gcn_combine_kernel<<<bN32, T, 0, stream>>>(agg, bufH, dinv, b2,
                                               (float*)d_out, N, /*relu=*/0);
}